// PagedAttention_33749853012079
// MI455X (gfx1250) — compile-verified
//
#include <hip/hip_runtime.h>
#include <math.h>

#define NUM_HEADS 8
#define HEAD 128
#define NUM_PROMPTS 4
#define PROMPT_LEN 512
#define NUM_GEN 64
#define KVBLOCK 16
#define MAX_CTX 2048
#define MAX_BLOCKS (MAX_CTX / KVBLOCK)      /* 128 */
#define NUM_CACHE_BLOCKS 8704
#define NP_TOK (NUM_PROMPTS * PROMPT_LEN)   /* 2048 */
#define NV_TOK (NP_TOK + NUM_GEN)           /* 2112 */
#define QSTRIDE (NUM_HEADS * HEAD)          /* 1024 */
#define SCALE 0.08838834764831845f          /* 1/sqrt(128) */

typedef float v2f __attribute__((ext_vector_type(2)));
typedef float v8f __attribute__((ext_vector_type(8)));

// ---------------------------------------------------------------------------
// Inverse slot map: inv[slot] = token index written there by slot_mapping, -1 otherwise.
// Lets the decode gather read fresh K/V from the inputs instead of needing a
// mutated copy of the 570MB caches.
// ---------------------------------------------------------------------------
__global__ void init_inv_kernel(int* __restrict__ inv, int n) {
    int i = blockIdx.x * blockDim.x + threadIdx.x;
    if (i < n) inv[i] = -1;
}

__global__ void scatter_inv_kernel(const int* __restrict__ slot_mapping,
                                   int* __restrict__ inv, int n) {
    int i = blockIdx.x * blockDim.x + threadIdx.x;
    if (i < n) inv[slot_mapping[i]] = i;
}

// ---------------------------------------------------------------------------
// Prompt phase: flash attention, fp32 WMMA (v_wmma_f32_16x16x4_f32).
// Block = 128 threads = 4 waves. Wave w owns q-tile (g*4 + w); the block
// shares K/V 16x128 tiles staged in LDS. Grid = 4 prompts * 8 heads * 8 groups.
// ---------------------------------------------------------------------------
__launch_bounds__(128)
__global__ void prompt_attn_kernel(const float* __restrict__ q,
                                   const float* __restrict__ k,
                                   const float* __restrict__ v,
                                   float* __restrict__ out)
{
    __shared__ float sQ[4][16][132];   // padded: 132 mod 64-banks = 4 -> conflict-free
    __shared__ float sK[16][132];
    __shared__ float sV[16][132];
    __shared__ float sP[4][16][20];    // padded: 20 -> conflict-free A-layout reads

    const int bid = blockIdx.x;
    const int b  = bid >> 6;           // prompt
    const int h  = (bid >> 3) & 7;     // head
    const int g  = bid & 7;            // q-group (4 tiles of 16)

    const int tid  = threadIdx.x;
    const int w    = tid >> 5;         // wave 0..3
    const int lane = tid & 31;
    const int m16  = lane & 15;        // row (A) / col (B,C,D) index
    const int hi   = lane >> 4;        // half-wave select
    const int koff = hi * 2;           // K-slice within WMMA step

    const int qt    = g * 4 + w;       // this wave's 16-row query tile
    const int ktmax = g * 4 + 3;       // causal bound for the whole block

    // ---- preload the 4 Q tiles (64 rows x 128) into LDS, float4 coalesced ----
    for (int it = 0; it < 16; ++it) {
        int idx = tid + it * 128;      // 0..2047 float4 chunks
        int row = idx >> 5;            // 0..63
        int c4  = idx & 31;
        int wv  = row >> 4;
        int r   = row & 15;
        int tok = b * PROMPT_LEN + (g * 4 + wv) * 16 + r;
        float4 val = *(const float4*)(q + (size_t)tok * QSTRIDE + h * HEAD + c4 * 4);
        *(float4*)&sQ[wv][r][c4 * 4] = val;   // row stride 528B -> 16B aligned
    }

    v8f acc[8];
    float m_run[8], l_run[8];
#pragma unroll
    for (int j = 0; j < 8; ++j) acc[j] = v8f{0.f,0.f,0.f,0.f,0.f,0.f,0.f,0.f};
#pragma unroll
    for (int r = 0; r < 8; ++r) { m_run[r] = -__builtin_inff(); l_run[r] = 0.f; }

    for (int kt = 0; kt <= ktmax; ++kt) {
        // ---- stage K/V tile (16x128 each) cooperatively ----
        for (int it = 0; it < 4; ++it) {
            int idx = tid + it * 128;  // 0..511 float4 chunks
            int row = idx >> 5;
            int c4  = idx & 31;
            int tok = b * PROMPT_LEN + kt * 16 + row;
            *(float4*)&sK[row][c4 * 4] =
                *(const float4*)(k + (size_t)tok * QSTRIDE + h * HEAD + c4 * 4);
            *(float4*)&sV[row][c4 * 4] =
                *(const float4*)(v + (size_t)tok * QSTRIDE + h * HEAD + c4 * 4);
        }
        __syncthreads();

        if (kt <= qt) {  // wave-uniform: EXEC stays all-ones for WMMA
            // ---- S = Q * K^T over d=0..127 in 32 fp32 WMMA steps ----
            v8f s = v8f{0.f,0.f,0.f,0.f,0.f,0.f,0.f,0.f};
#pragma unroll
            for (int d0 = 0; d0 < HEAD; d0 += 4) {
                v2f a  = *(const v2f*)&sQ[w][m16][d0 + koff];   // A[m][d0+koff..+1]
                v2f bb = *(const v2f*)&sK[m16][d0 + koff];      // B[kk][n]=K[n][d0+kk]
                s = __builtin_amdgcn_wmma_f32_16x16x4_f32(
                        false, a, false, bb, (short)0, s, false, false);
            }

            // ---- scale + causal mask (diagonal tile only) ----
#pragma unroll
            for (int r = 0; r < 8; ++r) {
                float sv_ = s[r] * SCALE;
                int row = r + hi * 8;
                if (kt == qt && m16 > row) sv_ = -__builtin_inff();
                s[r] = sv_;
            }

            // ---- online softmax (row stats via 16-lane shfl reductions) ----
#pragma unroll
            for (int r = 0; r < 8; ++r) {
                float mx = s[r];
                mx = fmaxf(mx, __shfl_xor(mx, 1));
                mx = fmaxf(mx, __shfl_xor(mx, 2));
                mx = fmaxf(mx, __shfl_xor(mx, 4));
                mx = fmaxf(mx, __shfl_xor(mx, 8));
                float nm = fmaxf(m_run[r], mx);
                float p  = __expf(s[r] - nm);
                float rs = p;
                rs += __shfl_xor(rs, 1);
                rs += __shfl_xor(rs, 2);
                rs += __shfl_xor(rs, 4);
                rs += __shfl_xor(rs, 8);
                float corr = __expf(m_run[r] - nm);
                l_run[r] = l_run[r] * corr + rs;
                m_run[r] = nm;
#pragma unroll
                for (int j = 0; j < 8; ++j) acc[j][r] *= corr;
                sP[w][r + hi * 8][m16] = p;   // C-layout -> LDS for A-layout reuse
            }
            // wave-private LDS RAW: LDS is in-order per wave; wait + block compiler motion
            asm volatile("s_wait_dscnt 0" ::: "memory");

            // ---- O += P * V : 8 column tiles x 4 WMMA steps ----
#pragma unroll
            for (int j = 0; j < 8; ++j) {
                v8f o = acc[j];
#pragma unroll
                for (int kc = 0; kc < 16; kc += 4) {
                    v2f a = *(const v2f*)&sP[w][m16][kc + koff];  // A[m][kc+koff..+1]
                    v2f bb;
                    bb.x = sV[kc + koff + 0][j * 16 + m16];       // B[kk][n]=V[kc+kk][n0+n]
                    bb.y = sV[kc + koff + 1][j * 16 + m16];
                    o = __builtin_amdgcn_wmma_f32_16x16x4_f32(
                            false, a, false, bb, (short)0, o, false, false);
                }
                acc[j] = o;
            }
        }
        __syncthreads();
    }

    // ---- normalize + write 16x128 output tile ----
#pragma unroll
    for (int r = 0; r < 8; ++r) {
        float invl = 1.0f / l_run[r];
        int qrow = b * PROMPT_LEN + qt * 16 + r + hi * 8;
#pragma unroll
        for (int j = 0; j < 8; ++j)
            out[(size_t)qrow * QSTRIDE + h * HEAD + j * 16 + m16] = acc[j][r] * invl;
    }
}

// ---------------------------------------------------------------------------
// Decode phase: one block per (gen token, head). Pure gather-bandwidth bound
// (~1GB of K/V traffic total) -> coalesced float4 gathers, scores in LDS.
// ---------------------------------------------------------------------------
__launch_bounds__(256)
__global__ void gen_attn_kernel(const float* __restrict__ q,
                                const float* __restrict__ k,
                                const float* __restrict__ v,
                                const float* __restrict__ kcache,
                                const float* __restrict__ vcache,
                                const int* __restrict__ block_tables,
                                const int* __restrict__ ctx_lens,
                                const int* __restrict__ inv,
                                float* __restrict__ out)
{
    __shared__ float sQg[HEAD];
    __shared__ float sS[MAX_CTX];
    __shared__ float sRed[16];
    __shared__ float sO[HEAD];

    const int mtok = blockIdx.x >> 3;
    const int h    = blockIdx.x & 7;
    const int tid  = threadIdx.x;
    const int wv   = tid >> 5;
    const int lane = tid & 31;
    const int ctx  = ctx_lens[mtok];

    if (tid < HEAD)
        sQg[tid] = q[(size_t)(NP_TOK + mtok) * QSTRIDE + h * HEAD + tid];
    __syncthreads();

    // ---- phase 1: scores. One wave per position, lane covers 4 dims ----
    float4 qv = *(const float4*)&sQg[lane * 4];
    for (int p = wv; p < ctx; p += 8) {
        int blk  = block_tables[mtok * MAX_BLOCKS + (p >> 4)];
        int slot = blk * KVBLOCK + (p & 15);
        int tok  = inv[slot];
        const float* kp = (tok >= 0)
            ? (k + (size_t)tok * QSTRIDE + h * HEAD)
            : (kcache + ((size_t)slot * NUM_HEADS + h) * HEAD);
        float4 kv4 = *(const float4*)(kp + lane * 4);
        float part = qv.x * kv4.x + qv.y * kv4.y + qv.z * kv4.z + qv.w * kv4.w;
        part += __shfl_xor(part, 16);
        part += __shfl_xor(part, 8);
        part += __shfl_xor(part, 4);
        part += __shfl_xor(part, 2);
        part += __shfl_xor(part, 1);
        if (lane == 0) sS[p] = part * SCALE;
    }
    __syncthreads();

    // ---- phase 2: softmax over sS[0..ctx) ----
    float mx = -__builtin_inff();
    for (int i = tid; i < ctx; i += 256) mx = fmaxf(mx, sS[i]);
    mx = fmaxf(mx, __shfl_xor(mx, 16));
    mx = fmaxf(mx, __shfl_xor(mx, 8));
    mx = fmaxf(mx, __shfl_xor(mx, 4));
    mx = fmaxf(mx, __shfl_xor(mx, 2));
    mx = fmaxf(mx, __shfl_xor(mx, 1));
    if (lane == 0) sRed[wv] = mx;
    __syncthreads();
    float bm = sRed[0];
#pragma unroll
    for (int j = 1; j < 8; ++j) bm = fmaxf(bm, sRed[j]);

    float lsum = 0.f;
    for (int i = tid; i < ctx; i += 256) {
        float e = __expf(sS[i] - bm);
        sS[i] = e;
        lsum += e;
    }
    lsum += __shfl_xor(lsum, 16);
    lsum += __shfl_xor(lsum, 8);
    lsum += __shfl_xor(lsum, 4);
    lsum += __shfl_xor(lsum, 2);
    lsum += __shfl_xor(lsum, 1);
    if (lane == 0) sRed[8 + wv] = lsum;
    __syncthreads();
    float lt = 0.f;
#pragma unroll
    for (int j = 0; j < 8; ++j) lt += sRed[8 + j];
    float invl = 1.0f / lt;

    // ---- phase 3: O = P * V. 128 dim-owners x 2 position phases ----
    const int d    = tid & 127;
    const int half = tid >> 7;
    float acc = 0.f;
    for (int p = half; p < ctx; p += 2) {
        int blk  = block_tables[mtok * MAX_BLOCKS + (p >> 4)];
        int slot = blk * KVBLOCK + (p & 15);
        int tok  = inv[slot];
        const float* vp = (tok >= 0)
            ? (v + (size_t)tok * QSTRIDE + h * HEAD)
            : (vcache + ((size_t)slot * NUM_HEADS + h) * HEAD);
        acc += sS[p] * vp[d];
    }
    if (half == 0) sO[d] = acc;
    __syncthreads();
    if (half == 1) sO[d] += acc;
    __syncthreads();
    if (tid < HEAD)
        out[(size_t)(NP_TOK + mtok) * QSTRIDE + h * HEAD + tid] = sO[tid] * invl;
}

// ---------------------------------------------------------------------------
extern "C" void kernel_launch(void* const* d_in, const int* in_sizes, int n_in,
                              void* d_out, int out_size, void* d_ws, size_t ws_size,
                              hipStream_t stream) {
    const float* query  = (const float*)d_in[0];
    const float* key    = (const float*)d_in[1];
    const float* value  = (const float*)d_in[2];
    const float* kcache = (const float*)d_in[3];
    const float* vcache = (const float*)d_in[4];
    const int* slot_mapping = (const int*)d_in[5];
    const int* block_tables = (const int*)d_in[6];
    const int* context_lens = (const int*)d_in[7];
    float* out = (float*)d_out;

    int* inv = (int*)d_ws;                         // 8704*16 ints = 544KB scratch
    const int nslots = NUM_CACHE_BLOCKS * KVBLOCK;

    init_inv_kernel<<<(nslots + 255) / 256, 256, 0, stream>>>(inv, nslots);
    scatter_inv_kernel<<<(NV_TOK + 255) / 256, 256, 0, stream>>>(slot_mapping, inv, NV_TOK);
    prompt_attn_kernel<<<NUM_PROMPTS * NUM_HEADS * 8, 128, 0, stream>>>(query, key, value, out);
    gen_attn_kernel<<<NUM_GEN * NUM_HEADS, 256, 0, stream>>>(
        query, key, value, kcache, vcache, block_tables, context_lens, inv, out);
}